// CVAModule_79826262163988
// MI455X (gfx1250) — compile-verified
//
#include <hip/hip_runtime.h>
#include <math.h>

// ---------------------------------------------------------------------------
// Deformable cross-attention for MI455X (gfx1250, wave32, WMMA).
// Matrix ops use V_WMMA_F32_16X16X4_F32 (fp32 in/out, no precision loss).
// Attention is single-pass per 16-row strip: S strip lives in LDS (64KB),
// softmax via half-wave shfl reductions, attn written to d_out exactly once.
// ---------------------------------------------------------------------------

typedef __attribute__((ext_vector_type(2))) float v2f;
typedef __attribute__((ext_vector_type(8))) float v8f;

#define WMMA_F32(a, b, c) \
  __builtin_amdgcn_wmma_f32_16x16x4_f32(false, (a), false, (b), (short)0, (c), false, false)

constexpr int B_ = 8;
constexpr int C_ = 384;
constexpr int H_ = 32;
constexpr int W_ = 32;
constexpr int N_ = 1024;          // H*W
constexpr int GROUPS_ = 3;
constexpr int CG_ = 128;          // C/GROUPS
constexpr int HEADS_ = 12;
constexpr int HC_ = 32;           // C/HEADS
constexpr size_t BCN_ = (size_t)B_ * C_ * N_;   // 3,145,728

// ---------------------------------------------------------------------------
// Generic batched projection GEMM:  Y[b,m,n] = sum_k Wt[m,k]*X[b,k,n] + bias[m]
// (optional residual add). Block = 4 waves, each wave one 16x16 WMMA tile,
// block tile = 32x32. K stepped by 4 per the f32 WMMA shape.
// A fragment (ISA 16x4 layout): lanes 0-15 -> K=k,k+1 ; lanes 16-31 -> K=k+2,k+3
// ---------------------------------------------------------------------------
__global__ __launch_bounds__(128)
void gemm_wmma(const float* __restrict__ Wt, const float* __restrict__ bias,
               const float* __restrict__ X, const float* __restrict__ res,
               float* __restrict__ Y, int M, int K, int NN)
{
  const int lane = threadIdx.x & 31;
  const int wid  = threadIdx.x >> 5;
  const int hl   = lane >> 4;       // half-wave select (K offset pair)
  const int l15  = lane & 15;
  const int m0 = (blockIdx.y * 2 + (wid >> 1)) * 16;
  const int n0 = (blockIdx.x * 2 + (wid & 1)) * 16;
  const size_t b = blockIdx.z;
  const float* Xb = X + b * (size_t)K * NN;

  v8f acc = {};
  for (int k = 0; k < K; k += 4) {
    const int ks = k + hl * 2;
    v2f a, bf;
    a.x  = Wt[(size_t)(m0 + l15) * K + ks];
    a.y  = Wt[(size_t)(m0 + l15) * K + ks + 1];
    bf.x = Xb[(size_t)ks * NN + n0 + l15];
    bf.y = Xb[(size_t)(ks + 1) * NN + n0 + l15];
    acc = WMMA_F32(a, bf, acc);
  }

  float* Yb = Y + b * (size_t)M * NN;
#pragma unroll
  for (int r = 0; r < 8; ++r) {        // C/D layout: VGPR r -> row r (+8 for hi half)
    const int m = m0 + r + hl * 8;
    const int n = n0 + l15;
    float val = acc[r] + bias[m];
    if (res) val += res[b * (size_t)M * NN + (size_t)m * NN + n];
    Yb[(size_t)m * NN + n] = val;
  }
}

// ---------------------------------------------------------------------------
// Offset prediction: depthwise 5x5 conv -> GELU(tanh) -> pointwise -> tanh
// -> clipped sampling grid. One thread per (bg, y, x).
// ---------------------------------------------------------------------------
__device__ __forceinline__ float gelu_tanh(float x) {
  return 0.5f * x * (1.0f + tanhf(0.7978845608028654f * (x + 0.044715f * x * x * x)));
}

__global__ __launch_bounds__(128)
void offset_grid_kernel(const float* __restrict__ q,
                        const float* __restrict__ w_dw, const float* __restrict__ b_dw,
                        const float* __restrict__ w_pw, const float* __restrict__ b_pw,
                        float* __restrict__ grid)
{
  const int idx = blockIdx.x * blockDim.x + threadIdx.x;
  if (idx >= B_ * GROUPS_ * N_) return;
  const int p  = idx & (N_ - 1);
  const int bg = idx >> 10;
  const int x = p & 31, y = p >> 5;
  const int b = bg / GROUPS_, g = bg % GROUPS_;

  float off0 = b_pw[0], off1 = b_pw[1];
  for (int cg = 0; cg < CG_; ++cg) {
    const float* qc = q + ((size_t)b * C_ + g * CG_ + cg) * N_;
    const float* wd = w_dw + cg * 25;
    float s = b_dw[cg];
    for (int ky = 0; ky < 5; ++ky) {
      const int yy = y + ky - 2;
      if (yy < 0 || yy >= H_) continue;
      for (int kx = 0; kx < 5; ++kx) {
        const int xx = x + kx - 2;
        if (xx < 0 || xx >= W_) continue;
        s += wd[ky * 5 + kx] * qc[yy * W_ + xx];
      }
    }
    const float u = gelu_tanh(s);
    off0 += w_pw[cg] * u;          // w_pw row 0
    off1 += w_pw[CG_ + cg] * u;    // w_pw row 1
  }
  const float dx = tanhf(off0) * (2.0f / W_);
  const float dy = tanhf(off1) * (2.0f / H_);
  const float rx = ((float)x + 0.5f) / W_ * 2.0f - 1.0f;
  const float ry = ((float)y + 0.5f) / H_ * 2.0f - 1.0f;
  grid[(size_t)idx * 2]     = fminf(fmaxf(rx + dx, -1.0f), 1.0f);
  grid[(size_t)idx * 2 + 1] = fminf(fmaxf(ry + dy, -1.0f), 1.0f);
}

// ---------------------------------------------------------------------------
// Bilinear sampling of x2 at the deformed grid (align_corners=True, clamped).
// One thread per (bg, cg, n).
// ---------------------------------------------------------------------------
__global__ __launch_bounds__(256)
void sample_kernel(const float* __restrict__ x2, const float* __restrict__ grid,
                   float* __restrict__ sampled)
{
  const size_t idx = (size_t)blockIdx.x * blockDim.x + threadIdx.x;
  if (idx >= BCN_) return;
  const int n  = (int)(idx & (N_ - 1));
  const int cg = (int)((idx >> 10) & (CG_ - 1));
  const int bg = (int)(idx >> 17);             // CG_*N_ = 2^17
  const int b = bg / GROUPS_, g = bg % GROUPS_;

  const float px = grid[((size_t)bg * N_ + n) * 2];
  const float py = grid[((size_t)bg * N_ + n) * 2 + 1];
  const float gx = (px + 1.0f) * 0.5f * (W_ - 1);
  const float gy = (py + 1.0f) * 0.5f * (H_ - 1);
  const float x0 = floorf(gx), y0 = floorf(gy);
  const float wx1 = gx - x0, wx0 = 1.0f - wx1;
  const float wy1 = gy - y0, wy0 = 1.0f - wy1;
  const int x0i = min(max((int)x0, 0), W_ - 1);
  const int x1i = min(max((int)x0 + 1, 0), W_ - 1);
  const int y0i = min(max((int)y0, 0), H_ - 1);
  const int y1i = min(max((int)y0 + 1, 0), H_ - 1);
  const float* img = x2 + ((size_t)b * C_ + g * CG_ + cg) * N_;
  const float v = wy0 * wx0 * img[y0i * W_ + x0i] + wy0 * wx1 * img[y0i * W_ + x1i]
                + wy1 * wx0 * img[y1i * W_ + x0i] + wy1 * wx1 * img[y1i * W_ + x1i];
  sampled[((size_t)b * C_ + g * CG_ + cg) * N_ + n] = v;
}

// ---------------------------------------------------------------------------
// v (B,C,N) -> vT (B,HEADS,N,HC) so AV B-fragments load lane-contiguously.
// ---------------------------------------------------------------------------
__global__ __launch_bounds__(256)
void transpose_v(const float* __restrict__ v, float* __restrict__ vT)
{
  const size_t idx = (size_t)blockIdx.x * blockDim.x + threadIdx.x;
  if (idx >= BCN_) return;
  const int n = (int)(idx & (N_ - 1));
  const size_t rest = idx >> 10;
  const int c = (int)(rest % C_);
  const int b = (int)(rest / C_);
  const int h = c / HC_, cc = c % HC_;
  vT[(((size_t)b * HEADS_ + h) * N_ + n) * HC_ + cc] = v[idx];
}

// ---------------------------------------------------------------------------
// Fused attention: one workgroup per (b, h, 16-row strip of n).
//  Phase 1 (8 waves): S[16,1024] = scale * q^T k  -> LDS (XOR-swizzled banks);
//                     the Q A-fragment is hoisted out of the m-tile loop.
//  Phase 2 (256 thr): row softmax via half-wave shfl; write attn to d_out
//  Phase 3 (8 waves): out = attn @ V^T; 4 waves per cc-half each reduce a
//                     256-m quarter, partial tiles summed through LDS (S reused)
// ---------------------------------------------------------------------------
#define SIDX(r, m) ((m) ^ (((r) & 15) << 2))

__global__ __launch_bounds__(256)
void attn_kernel(const float* __restrict__ q, const float* __restrict__ k,
                 const float* __restrict__ vT, float* __restrict__ attn,
                 float* __restrict__ obuf)
{
  __shared__ float S[16][N_];     // 64 KB (of 320 KB per WGP)
  const int b  = blockIdx.z, h = blockIdx.y;
  const int n0 = blockIdx.x * 16;
  const int tid = threadIdx.x;
  const int lane = tid & 31, wid = tid >> 5;
  const int hl = lane >> 4, l15 = lane & 15;
  const float scale = 0.17677669529663687f;   // hc^-0.5
  const float* qh = q  + ((size_t)b * C_ + h * HC_) * N_;
  const float* kh = k  + ((size_t)b * C_ + h * HC_) * N_;
  const float* vh = vT + ((size_t)b * HEADS_ + h) * (size_t)N_ * HC_;

  // ---- Phase 1: logits strip; Q fragments loaded once, reused for 64 m-tiles
  v2f aq[8];
#pragma unroll
  for (int i = 0; i < 8; ++i) {
    const int ks = i * 4 + hl * 2;
    aq[i].x = qh[(size_t)ks * N_ + n0 + l15];        // A[n, cc] = q[cc, n]
    aq[i].y = qh[(size_t)(ks + 1) * N_ + n0 + l15];
  }
  for (int mt = wid; mt < N_ / 16; mt += 8) {
    const int m0 = mt * 16;
    v8f acc = {};
#pragma unroll
    for (int i = 0; i < 8; ++i) {
      const int ks = i * 4 + hl * 2;
      v2f bf;
      bf.x = kh[(size_t)ks * N_ + m0 + l15];         // B[cc, m] = k[cc, m]
      bf.y = kh[(size_t)(ks + 1) * N_ + m0 + l15];
      acc = WMMA_F32(aq[i], bf, acc);
    }
#pragma unroll
    for (int r = 0; r < 8; ++r) {
      const int rr = r + hl * 8;
      S[rr][SIDX(rr, m0 + l15)] = acc[r] * scale;
    }
  }
  __syncthreads();

  // ---- Phase 2: softmax per row, 16 threads per row, shfl reductions ----
  {
    const int row = tid >> 4, lir = tid & 15;
    float lmax = -1e30f;
    for (int m = lir; m < N_; m += 16) lmax = fmaxf(lmax, S[row][SIDX(row, m)]);
    for (int o = 1; o < 16; o <<= 1) lmax = fmaxf(lmax, __shfl_xor(lmax, o, 16));
    float lsum = 0.0f;
    for (int m = lir; m < N_; m += 16) {
      const float e = __expf(S[row][SIDX(row, m)] - lmax);
      S[row][SIDX(row, m)] = e;
      lsum += e;
    }
    for (int o = 1; o < 16; o <<= 1) lsum += __shfl_xor(lsum, o, 16);
    const float inv = 1.0f / lsum;
    float* arow = attn + (((size_t)b * HEADS_ + h) * N_ + (n0 + row)) * N_;
    for (int m = lir; m < N_; m += 16) {
      const float pr = S[row][SIDX(row, m)] * inv;
      S[row][SIDX(row, m)] = pr;
      arow[m] = pr;                                  // attn written exactly once
    }
  }
  __syncthreads();

  // ---- Phase 3: out = attn @ V^T over all 8 waves.
  // wave = quarter*2 + half : half = cc-half (16 ch), quarter = 256-m slice.
  v8f acc3 = {};
  {
    const int cc0 = (wid & 1) * 16;
    const int mq  = wid >> 1;                        // 0..3
    for (int kk = mq * 256; kk < mq * 256 + 256; kk += 4) {
      const int ks = kk + hl * 2;
      v2f a, bf;
      a.x  = S[l15][SIDX(l15, ks)];                  // A[n, m] from LDS (swizzled)
      a.y  = S[l15][SIDX(l15, ks + 1)];
      bf.x = vh[(size_t)ks * HC_ + cc0 + l15];       // B[m, cc] = vT[m, cc]
      bf.y = vh[(size_t)(ks + 1) * HC_ + cc0 + l15];
      acc3 = WMMA_F32(a, bf, acc3);
    }
  }
  __syncthreads();                 // all waves done READING S; safe to reuse it
  {
    float* psum = &S[0][0];        // 8 partial 16x16 tiles, 256 floats each
#pragma unroll
    for (int r = 0; r < 8; ++r) psum[wid * 256 + r * 32 + lane] = acc3[r];
  }
  __syncthreads();
  {
    const float* psum = &S[0][0];
    for (int e = tid; e < 512; e += 256) {           // 2 cc-halves x 256 elems
      const int half = e >> 8;
      const int elem = e & 255;
      const int r = elem >> 5, ln = elem & 31;
      const float s = psum[(0 * 2 + half) * 256 + elem]
                    + psum[(1 * 2 + half) * 256 + elem]
                    + psum[(2 * 2 + half) * 256 + elem]
                    + psum[(3 * 2 + half) * 256 + elem];
      const int n  = n0 + r + (ln >> 4) * 8;         // D layout row
      const int cc = half * 16 + (ln & 15);          // D layout col
      obuf[((size_t)b * C_ + h * HC_ + cc) * N_ + n] = s;
    }
  }
}

// ---------------------------------------------------------------------------
extern "C" void kernel_launch(void* const* d_in, const int* in_sizes, int n_in,
                              void* d_out, int out_size, void* d_ws, size_t ws_size,
                              hipStream_t stream) {
  const float* x1   = (const float*)d_in[0];
  const float* x2   = (const float*)d_in[1];
  const float* wq   = (const float*)d_in[2];
  const float* bq   = (const float*)d_in[3];
  const float* w_dw = (const float*)d_in[4];
  const float* b_dw = (const float*)d_in[5];
  const float* w_pw = (const float*)d_in[6];
  const float* b_pw = (const float*)d_in[7];
  const float* wk   = (const float*)d_in[8];
  const float* bk   = (const float*)d_in[9];
  const float* wv   = (const float*)d_in[10];
  const float* bv   = (const float*)d_in[11];
  const float* wo   = (const float*)d_in[12];
  const float* bo   = (const float*)d_in[13];

  float* ws      = (float*)d_ws;
  float* q       = ws;                  // (B,C,N)
  float* sampled = q + BCN_;            // (B,C,N)  (reused as obuf after k/v GEMMs)
  float* kbuf    = sampled + BCN_;      // (B,C,N)
  float* vbuf    = kbuf + BCN_;         // (B,C,N)
  float* vTbuf   = vbuf + BCN_;         // (B,HEADS,N,HC)
  float* gridbuf = vTbuf + BCN_;        // (B*G,N,2)
  float* obuf    = sampled;             // alias: sampled is dead after v GEMM

  float* y_out = (float*)d_out;                 // (B,C,H,W)
  float* attn  = (float*)d_out + BCN_;          // (B,HEADS,N,N)

  const dim3 gemm_grid(N_ / 32, C_ / 32, B_);   // (32, 12, 8)

  // q = wq @ x1 + bq
  gemm_wmma<<<gemm_grid, 128, 0, stream>>>(wq, bq, x1, nullptr, q, C_, C_, N_);
  // offsets -> sampling grid
  offset_grid_kernel<<<(B_ * GROUPS_ * N_ + 127) / 128, 128, 0, stream>>>(
      q, w_dw, b_dw, w_pw, b_pw, gridbuf);
  // bilinear sample x2 at deformed positions
  sample_kernel<<<(int)(BCN_ / 256), 256, 0, stream>>>(x2, gridbuf, sampled);
  // k/v projections
  gemm_wmma<<<gemm_grid, 128, 0, stream>>>(wk, bk, sampled, nullptr, kbuf, C_, C_, N_);
  gemm_wmma<<<gemm_grid, 128, 0, stream>>>(wv, bv, sampled, nullptr, vbuf, C_, C_, N_);
  // v -> vT for coalesced AV fragments
  transpose_v<<<(int)(BCN_ / 256), 256, 0, stream>>>(vbuf, vTbuf);
  // fused attention (logits + softmax + AV), writes attn + obuf
  attn_kernel<<<dim3(N_ / 16, HEADS_, B_), 256, 0, stream>>>(q, kbuf, vTbuf, attn, obuf);
  // y = wo @ out + bo + x1
  gemm_wmma<<<gemm_grid, 128, 0, stream>>>(wo, bo, obuf, x1, y_out, C_, C_, N_);
}